// WavefunctionEmbedding_67456756351126
// MI455X (gfx1250) — compile-verified
//
#include <hip/hip_runtime.h>
#include <hip/hip_bf16.h>
#include <math.h>
#include <stdint.h>

// ---------------------------------------------------------------------------
// WavefunctionEmbedding for MI455X (gfx1250, wave32)
//  Phase 1: kvec = 2*pi / (softmax(W) @ wl_buf)                 [K=256]
//  Phase 2: wavefield wf[b,i,:] = sum_j amp_ij * {cos,sin}(r_ij * k)
//           (coords staged to LDS via global_load_async_to_lds_b128)
//  Phase 3: h = gelu(wf @ w_in^T + b_in)    -- f32 WMMA, 32x32/wave
//  Phase 4: y = wf + h @ w_out^T + b_out    -- f32 WMMA, 32x32/wave
//  Phase 5: LayerNorm(y) * g + b
// ---------------------------------------------------------------------------

typedef __attribute__((ext_vector_type(2))) float v2f;
typedef __attribute__((ext_vector_type(8))) float v8f;

#define D_MODEL 512
#define KWAVE   256
#define D_HID   1024
#define NTOK    1024      // B*N = 2*512
#define NPTS    512

// ---------------------------------------------------------------------------
// Phase 1: per-row softmax over [256,256] mixed with wl_buf -> kvec[256]
// ---------------------------------------------------------------------------
__global__ void wfe_kvec_kernel(const float* __restrict__ ww,
                                const float* __restrict__ wl_buf,
                                float* __restrict__ kvec) {
    const int r = threadIdx.x;                 // 256 threads, one row each
    const float* row = ww + r * KWAVE;
    float mx = -3.4e38f;
    for (int j = 0; j < KWAVE; ++j) mx = fmaxf(mx, row[j]);
    float se = 0.0f, dot = 0.0f;
    for (int j = 0; j < KWAVE; ++j) {
        float e = __expf(row[j] - mx);
        se += e;
        dot = fmaf(e, wl_buf[j], dot);
    }
    kvec[r] = 6.2831853071795864f * se / dot;  // 2*pi / (dot/se)
}

// ---------------------------------------------------------------------------
// Phase 2: one block per (b,i); coords for the batch staged in LDS via
// CDNA5 async-to-LDS (ASYNCcnt-tracked, no VGPR round-trip), r_j/amp_j
// staged in LDS, each lane owns one wavenumber channel.
// Inner loop: v_sin/v_cos + FMA off LDS broadcasts.
// ---------------------------------------------------------------------------
__global__ void wfe_wavefield_kernel(const float* __restrict__ coords,
                                     const unsigned char* __restrict__ pad,
                                     const float* __restrict__ kvec,
                                     float* __restrict__ wf) {
    const int bi = blockIdx.x;                 // b*512 + i
    const int b  = bi >> 9;
    const int i  = bi & (NPTS - 1);

    __shared__ __align__(16) float scoord[NPTS * 3];   // 6 KB: batch coords
    __shared__ float sr[NPTS];
    __shared__ float samp[NPTS];

    const float* cbase = coords + (size_t)b * NPTS * 3;

    // 1536 floats = 384 x b128 async transfers. LDS dest address is the
    // wave-relative offset = low 32 bits of the generic shared pointer.
    for (int t = threadIdx.x; t < (NPTS * 3) / 4; t += 256) {
        const uint32_t lds_addr = (uint32_t)(uintptr_t)&scoord[4 * t];
        const uint64_t gaddr    = (uint64_t)(uintptr_t)(cbase + 4 * t);
        asm volatile("global_load_async_to_lds_b128 %0, %1, off"
                     :: "v"(lds_addr), "v"(gaddr) : "memory");
    }
    asm volatile("s_wait_asynccnt 0x0" ::: "memory");
    __syncthreads();

    const float xi = scoord[i * 3 + 0];
    const float yi = scoord[i * 3 + 1];
    const float zi = scoord[i * 3 + 2];

    for (int j = threadIdx.x; j < NPTS; j += blockDim.x) {
        const float dx = xi - scoord[j * 3 + 0];
        const float dy = yi - scoord[j * 3 + 1];
        const float dz = zi - scoord[j * 3 + 2];
        const float r2 = fmaf(dx, dx, fmaf(dy, dy, dz * dz));
        const bool valid = (j != i) && (pad[b * NPTS + j] == 0);
        const float r = sqrtf(valid ? r2 : 1.0f);
        sr[j]   = r;
        samp[j] = valid ? (0.07957747154594767f / r) : 0.0f;  // 1/(4*pi*r)
    }
    __syncthreads();

    const int k = threadIdx.x;                 // 256 threads == K channels
    const float kv = kvec[k];
    float re = 0.0f, im = 0.0f;
#pragma unroll 4
    for (int j = 0; j < NPTS; ++j) {
        const float a  = samp[j];
        const float ph = sr[j] * kv;
        float s, c;
        __sincosf(ph, &s, &c);
        re = fmaf(c, a, re);
        im = fmaf(s, a, im);
    }
    const float qv = (pad[bi] == 0) ? 1.0f : 0.0f;
    wf[bi * D_MODEL + k]         = re * qv;
    wf[bi * D_MODEL + KWAVE + k] = im * qv;
}

// ---------------------------------------------------------------------------
// f32 WMMA fragment layouts (ISA 7.12.2):
//  A 16x4 : a = { A[m0+col][k+2*half], A[m0+col][k+2*half+1] }
//  B 4x16 : b = { Bmat[k+2*half][n0+col], Bmat[k+2*half+1][n0+col] }
//           (Bmat[k][n] = W[n][k] row-major -> 2 consecutive floats)
//  D      : c[v] = D[m0 + 8*half + v][n0 + col]
// Each wave owns a 32x32 tile: 2 A frags x 2 B frags -> 4 accumulators,
// 4 independent WMMA chains per K-step from 4 x 8B loads.
// ---------------------------------------------------------------------------

// Phase 3: h[1024,1024] = gelu(wf[1024,512] @ w_in^T + b_in)
__global__ void wfe_gemm1_gelu_kernel(const float* __restrict__ wf,
                                      const float* __restrict__ w_in,
                                      const float* __restrict__ b_in,
                                      float* __restrict__ h) {
    const int lane = threadIdx.x & 31;
    const int wave = threadIdx.x >> 5;                 // 8 waves / block
    const int mt   = blockIdx.x >> 2;                  // 0..31 (M = 1024/32)
    const int nt   = ((blockIdx.x & 3) << 3) + wave;   // 0..31 (N = 1024/32)
    const int m0 = mt * 32, n0 = nt * 32;
    const int half = lane >> 4;
    const int col  = lane & 15;

    const float* Ap0 = wf   + (size_t)(m0 + col) * D_MODEL + half * 2;
    const float* Ap1 = Ap0  + 16 * D_MODEL;
    const float* Bp0 = w_in + (size_t)(n0 + col) * D_MODEL + half * 2;
    const float* Bp1 = Bp0  + 16 * D_MODEL;

    v8f c00 = {}, c01 = {}, c10 = {}, c11 = {};
#pragma unroll 4
    for (int k = 0; k < D_MODEL; k += 4) {
        const v2f a0 = *(const v2f*)(Ap0 + k);
        const v2f a1 = *(const v2f*)(Ap1 + k);
        const v2f b0 = *(const v2f*)(Bp0 + k);
        const v2f b1 = *(const v2f*)(Bp1 + k);
        c00 = __builtin_amdgcn_wmma_f32_16x16x4_f32(false, a0, false, b0, (short)0, c00, false, false);
        c01 = __builtin_amdgcn_wmma_f32_16x16x4_f32(false, a0, false, b1, (short)0, c01, false, false);
        c10 = __builtin_amdgcn_wmma_f32_16x16x4_f32(false, a1, false, b0, (short)0, c10, false, false);
        c11 = __builtin_amdgcn_wmma_f32_16x16x4_f32(false, a1, false, b1, (short)0, c11, false, false);
    }

    const float bias0 = b_in[n0 + col];
    const float bias1 = b_in[n0 + 16 + col];
    float* out00 = h + (size_t)(m0 + half * 8) * D_HID + n0 + col;
#pragma unroll
    for (int v = 0; v < 8; ++v) {
        const float x0 = c00[v] + bias0;
        const float x1 = c01[v] + bias1;
        const float x2 = c10[v] + bias0;
        const float x3 = c11[v] + bias1;
        out00[(size_t)v * D_HID]             = 0.5f * x0 * (1.0f + erff(x0 * 0.70710678118654752f));
        out00[(size_t)v * D_HID + 16]        = 0.5f * x1 * (1.0f + erff(x1 * 0.70710678118654752f));
        out00[(size_t)(v + 16) * D_HID]      = 0.5f * x2 * (1.0f + erff(x2 * 0.70710678118654752f));
        out00[(size_t)(v + 16) * D_HID + 16] = 0.5f * x3 * (1.0f + erff(x3 * 0.70710678118654752f));
    }
}

// Phase 4: y[1024,512] = wf + h[1024,1024] @ w_out^T + b_out
__global__ void wfe_gemm2_res_kernel(const float* __restrict__ h,
                                     const float* __restrict__ w_out,
                                     const float* __restrict__ b_out,
                                     const float* __restrict__ wf,
                                     float* __restrict__ y) {
    const int lane = threadIdx.x & 31;
    const int wave = threadIdx.x >> 5;
    const int mt   = blockIdx.x >> 1;                  // 0..31 (M = 1024/32)
    const int nt   = ((blockIdx.x & 1) << 3) + wave;   // 0..15 (N = 512/32)
    const int m0 = mt * 32, n0 = nt * 32;
    const int half = lane >> 4;
    const int col  = lane & 15;

    const float* Ap0 = h     + (size_t)(m0 + col) * D_HID + half * 2;
    const float* Ap1 = Ap0   + 16 * D_HID;
    const float* Bp0 = w_out + (size_t)(n0 + col) * D_HID + half * 2;
    const float* Bp1 = Bp0   + 16 * D_HID;

    v8f c00 = {}, c01 = {}, c10 = {}, c11 = {};
#pragma unroll 4
    for (int k = 0; k < D_HID; k += 4) {
        const v2f a0 = *(const v2f*)(Ap0 + k);
        const v2f a1 = *(const v2f*)(Ap1 + k);
        const v2f b0 = *(const v2f*)(Bp0 + k);
        const v2f b1 = *(const v2f*)(Bp1 + k);
        c00 = __builtin_amdgcn_wmma_f32_16x16x4_f32(false, a0, false, b0, (short)0, c00, false, false);
        c01 = __builtin_amdgcn_wmma_f32_16x16x4_f32(false, a0, false, b1, (short)0, c01, false, false);
        c10 = __builtin_amdgcn_wmma_f32_16x16x4_f32(false, a1, false, b0, (short)0, c10, false, false);
        c11 = __builtin_amdgcn_wmma_f32_16x16x4_f32(false, a1, false, b1, (short)0, c11, false, false);
    }

    const float bias0 = b_out[n0 + col];
    const float bias1 = b_out[n0 + 16 + col];
    const size_t base = (size_t)(m0 + half * 8) * D_MODEL + n0 + col;
#pragma unroll
    for (int v = 0; v < 8; ++v) {
        const size_t i00 = base + (size_t)v * D_MODEL;
        const size_t i01 = i00 + 16;
        const size_t i10 = base + (size_t)(v + 16) * D_MODEL;
        const size_t i11 = i10 + 16;
        y[i00] = c00[v] + bias0 + wf[i00];
        y[i01] = c01[v] + bias1 + wf[i01];
        y[i10] = c10[v] + bias0 + wf[i10];
        y[i11] = c11[v] + bias1 + wf[i11];
    }
}

// ---------------------------------------------------------------------------
// Phase 5: row-wise LayerNorm over D_MODEL=512; 256 threads, 2 elems each.
// ---------------------------------------------------------------------------
__global__ void wfe_layernorm_kernel(const float* __restrict__ y,
                                     const float* __restrict__ g,
                                     const float* __restrict__ beta,
                                     float* __restrict__ out) {
    const int row = blockIdx.x;
    const int tid = threadIdx.x;               // 256 threads
    __shared__ float s1[256];
    __shared__ float s2[256];

    const float* x = y + (size_t)row * D_MODEL;
    const float a = x[tid];
    const float b = x[tid + 256];
    s1[tid] = a + b;
    s2[tid] = fmaf(a, a, b * b);
    __syncthreads();

    for (int off = 128; off > 0; off >>= 1) {
        if (tid < off) {
            s1[tid] += s1[tid + off];
            s2[tid] += s2[tid + off];
        }
        __syncthreads();
    }

    const float mu  = s1[0] * (1.0f / D_MODEL);
    const float var = s2[0] * (1.0f / D_MODEL) - mu * mu;
    const float inv = rsqrtf(var + 1e-5f);

    float* o = out + (size_t)row * D_MODEL;
    o[tid]       = (a - mu) * inv * g[tid]       + beta[tid];
    o[tid + 256] = (b - mu) * inv * g[tid + 256] + beta[tid + 256];
}

// ---------------------------------------------------------------------------
extern "C" void kernel_launch(void* const* d_in, const int* in_sizes, int n_in,
                              void* d_out, int out_size, void* d_ws, size_t ws_size,
                              hipStream_t stream) {
    const float*         coords = (const float*)d_in[0];
    const unsigned char* pad    = (const unsigned char*)d_in[1];  // numpy bool
    const float*         ww     = (const float*)d_in[2];
    const float*         wl_buf = (const float*)d_in[3];
    const float*         w_in   = (const float*)d_in[4];
    const float*         b_in   = (const float*)d_in[5];
    const float*         w_out  = (const float*)d_in[6];
    const float*         b_out  = (const float*)d_in[7];
    const float*         ln_g   = (const float*)d_in[8];
    const float*         ln_b   = (const float*)d_in[9];
    float* out = (float*)d_out;

    float* ws   = (float*)d_ws;
    float* kvec = ws;                          // 256
    float* wf   = kvec + KWAVE;                // 1024*512
    float* h    = wf + (size_t)NTOK * D_MODEL; // 1024*1024
    float* y    = h  + (size_t)NTOK * D_HID;   // 1024*512

    wfe_kvec_kernel<<<1, 256, 0, stream>>>(ww, wl_buf, kvec);
    wfe_wavefield_kernel<<<NTOK, 256, 0, stream>>>(coords, pad, kvec, wf);
    wfe_gemm1_gelu_kernel<<<128, 256, 0, stream>>>(wf, w_in, b_in, h);     // 32x32 tiles, 8 waves/blk
    wfe_gemm2_res_kernel<<<64, 256, 0, stream>>>(h, w_out, b_out, wf, y);  // 32x32 tiles, 8 waves/blk
    wfe_layernorm_kernel<<<NTOK, 256, 0, stream>>>(y, ln_g, ln_b, out);
}